// NeighbourLoss_37082747634119
// MI455X (gfx1250) — compile-verified
//
#include <hip/hip_runtime.h>
#include <math.h>

// MI455X / gfx1250 (CDNA5, wave32). Fused NeighbourLoss:
//   dist = sqrt(clip(||xi||^2+||xj||^2-2 xi.xj, 1e-12)), Gram via V_WMMA_F32_16X16X4_F32.
// One workgroup owns a 16-row slab; the full 16xN distance slab lives in LDS
// (320KB/WGP) so the pos_min -> hard-negative two-pass never touches HBM.
// Hot kernel is templated on D so the K-loop is a compile-time trip count
// (no scalar remainder loop, deeper unroll of the load->WMMA pipeline).

typedef __attribute__((ext_vector_type(2))) float v2f;
typedef __attribute__((ext_vector_type(8))) float v8f;

#define EPSF    1e-12f
#define MARGINF 0.1f

// ---------------------------------------------------------------------------
// Kernel 1: row squared-norms (X is 8MB, trivially bandwidth-bound) + zero the
// six global accumulators each call (harness does not re-zero ws).
// ---------------------------------------------------------------------------
__global__ void sq_norm_kernel(const float* __restrict__ X, float* __restrict__ sq,
                               float* __restrict__ acc, int N, int D) {
  int r = blockIdx.x * blockDim.x + threadIdx.x;
  if (r < 8) acc[r] = 0.0f;
  if (r >= N) return;
  const float4* p = reinterpret_cast<const float4*>(X + (size_t)r * D);
  float s = 0.f;
  int nd4 = D >> 2;
  for (int i = 0; i < nd4; ++i) {
    float4 v = p[i];
    s += v.x * v.x + v.y * v.y + v.z * v.z + v.w * v.w;
  }
  sq[r] = s;
}

// ---------------------------------------------------------------------------
// Kernel 2: per-16-row-block fused Gram (WMMA f32 16x16x4) + distance +
// masked reductions; distance slab held in LDS for the second pass.
//   LDS: Abuf 16x(DD+4) | distbuf 16x(N+4) | tgtcol N | per-row stats
// blockDim = 512 (16 waves); each wave computes two 16x16 tiles at a time.
// DD > 0: compile-time D (fully counted K-loop, no remainder). DD == 0: generic.
// ---------------------------------------------------------------------------
template <int DD>
__global__ void __launch_bounds__(512)
neighbour_main(const float* __restrict__ X, const int* __restrict__ targets,
               const float* __restrict__ sq, float* __restrict__ acc,
               int N, int Ddyn) {
  const int D   = (DD > 0) ? DD : Ddyn;
  extern __shared__ char smem_raw[];
  const int AST = D + 4;   // padded A row stride (floats): banks advance 4/row -> conflict-free b64
  const int DST = N + 4;   // padded dist row stride (floats): kills lane0-15 vs 16-31 conflicts
  float*    Abuf    = (float*)smem_raw;               // 16*AST floats
  float*    distbuf = Abuf + 16 * AST;                // 16*DST floats
  int*      tgtcol  = (int*)(distbuf + 16 * DST);     // N ints
  unsigned* posminb = (unsigned*)(tgtcol + N);        // 16 (float bits; nonneg -> uint order)
  float*    possum  = (float*)(posminb + 16);
  unsigned* poscnt  = (unsigned*)(possum + 16);
  float*    negsum  = (float*)(poscnt + 16);
  unsigned* negcnt  = (unsigned*)(negsum + 16);
  float*    hsum    = (float*)(negcnt + 16);
  unsigned* hcnt    = (unsigned*)(hsum + 16);
  float*    sqA     = (float*)(hcnt + 16);
  int*      tgtA    = (int*)(sqA + 16);

  const int tid = threadIdx.x;
  const int r0  = blockIdx.x * 16;

  // Stage the 16 x D A-panel into LDS (coalesced float4).
  {
    const int rowv = D >> 2;                 // float4s per row
    const int nvec = 16 * rowv;
    for (int idx = tid; idx < nvec; idx += blockDim.x) {
      int r  = idx / rowv;
      int c4 = idx - r * rowv;
      float4 v = reinterpret_cast<const float4*>(X + (size_t)(r0 + r) * D)[c4];
      float* dst = Abuf + r * AST + c4 * 4;
      dst[0] = v.x; dst[1] = v.y; dst[2] = v.z; dst[3] = v.w;
    }
  }
  for (int c = tid; c < N; c += blockDim.x) tgtcol[c] = targets[c];
  if (tid < 16) {
    sqA[tid]    = sq[r0 + tid];
    tgtA[tid]   = targets[r0 + tid];
    posminb[tid] = 0x7f800000u;  // +inf
    possum[tid] = 0.f; poscnt[tid] = 0u;
    negsum[tid] = 0.f; negcnt[tid] = 0u;
    hsum[tid]   = 0.f; hcnt[tid]   = 0u;
  }
  __syncthreads();

  const int wave   = tid >> 5;
  const int lane   = tid & 31;
  const int lo     = lane & 15;   // A: M row / B: N col / C: N col
  const int hi     = lane >> 4;   // K-half selector (WMMA f32 16x16x4 layout)
  const int nwaves = blockDim.x >> 5;

  float sqAr[8]; int tAr[8];
#pragma unroll
  for (int v = 0; v < 8; ++v) { sqAr[v] = sqA[v + hi * 8]; tAr[v] = tgtA[v + hi * 8]; }

  float pmin[8], psum[8], nsum[8];
  int   pcn[8], ncn[8];
#pragma unroll
  for (int v = 0; v < 8; ++v) { pmin[v] = INFINITY; psum[v] = 0.f; nsum[v] = 0.f; pcn[v] = 0; ncn[v] = 0; }

  const float* pA = Abuf + lo * AST + hi * 2;  // A frag: lane M=lo, K pair = hi*2 (+k)
  const int npairs = N >> 5;                   // column-tile pairs of 32 cols
  for (int tp = wave; tp < npairs; tp += nwaves) {   // wave-uniform: EXEC stays all-1s for WMMA
    const int c0a = tp * 32;
    const int c0b = c0a + 16;
    const float* pB0 = X + (size_t)(c0a + lo) * D + hi * 2;  // B frag: N=lo, K pair = hi*2
    const float* pB1 = X + (size_t)(c0b + lo) * D + hi * 2;
    v8f acc0 = {0.f, 0.f, 0.f, 0.f, 0.f, 0.f, 0.f, 0.f};
    v8f acc1 = acc0;
#pragma unroll 8
    for (int k = 0; k < D; k += 4) {
      v2f a  = *reinterpret_cast<const v2f*>(pA  + k);  // ds_load_b64 (pairs fuse to 2addr), conflict-free
      v2f b0 = *reinterpret_cast<const v2f*>(pB0 + k);  // global_load_b64, L2-resident
      v2f b1 = *reinterpret_cast<const v2f*>(pB1 + k);
      acc0 = __builtin_amdgcn_wmma_f32_16x16x4_f32(false, a, false, b0, (short)0, acc0, false, false);
      acc1 = __builtin_amdgcn_wmma_f32_16x16x4_f32(false, a, false, b1, (short)0, acc1, false, false);
    }
    union { v8f v; float f[8]; } u0, u1;
    u0.v = acc0; u1.v = acc1;
    float sqB0 = sq[c0a + lo], sqB1 = sq[c0b + lo];
    int   tB0  = tgtcol[c0a + lo], tB1 = tgtcol[c0b + lo];
#pragma unroll
    for (int v = 0; v < 8; ++v) {
      int M    = v + hi * 8;            // C layout: VGPR v -> row M (lo half) / M+8 (hi half)
      int rowg = r0 + M;
      float d0 = sqrtf(fmaxf(sqAr[v] + sqB0 - 2.f * u0.f[v], EPSF));
      float d1 = sqrtf(fmaxf(sqAr[v] + sqB1 - 2.f * u1.f[v], EPSF));
      distbuf[M * DST + c0a + lo] = d0;
      distbuf[M * DST + c0b + lo] = d1;
      bool same0 = (tAr[v] == tB0), same1 = (tAr[v] == tB1);
      if (same0) { if (rowg != c0a + lo) { pmin[v] = fminf(pmin[v], d0); psum[v] += d0; pcn[v]++; } }
      else       { nsum[v] += d0; ncn[v]++; }
      if (same1) { if (rowg != c0b + lo) { pmin[v] = fminf(pmin[v], d1); psum[v] += d1; pcn[v]++; } }
      else       { nsum[v] += d1; ncn[v]++; }
    }
  }
  // Fold per-lane partials into per-row LDS stats (ds_min via uint bits, ds_add_f32).
#pragma unroll
  for (int v = 0; v < 8; ++v) {
    int M = v + hi * 8;
    atomicMin(&posminb[M], __float_as_uint(pmin[v]));
    atomicAdd(&possum[M], psum[v]);
    atomicAdd(&poscnt[M], (unsigned)pcn[v]);
    atomicAdd(&negsum[M], nsum[v]);
    atomicAdd(&negcnt[M], (unsigned)ncn[v]);
  }
  __syncthreads();

  // Pass 2: hard negatives (needs pos_min of the full row). 32 threads per row,
  // consecutive columns -> conflict-free LDS reads from the resident dist slab.
  {
    int row = tid >> 5;            // 512 threads -> rows 0..15
    int c0  = tid & 31;
    float thr = __uint_as_float(posminb[row]) + MARGINF;
    int tr = tgtA[row];
    const float* drow = distbuf + row * DST;
    float hs = 0.f; int hc = 0;
#pragma unroll 4
    for (int c = c0; c < N; c += 32) {
      float d = drow[c];
      if ((tgtcol[c] != tr) && (d < thr)) { hs += d; hc++; }
    }
    atomicAdd(&hsum[row], hs);
    atomicAdd(&hcnt[row], (unsigned)hc);
  }
  __syncthreads();

  if (tid == 0) {  // one global atomic per statistic per block
    float loss_s = 0.f, err = 0.f, ps = 0.f, pc = 0.f, ns = 0.f, nc = 0.f;
    for (int r = 0; r < 16; ++r) {
      float pm = __uint_as_float(posminb[r]);
      unsigned h = hcnt[r];
      if (h > 0) { loss_s += pm - hsum[r] / (float)h + MARGINF; err += 1.f; }
      ps += possum[r]; pc += (float)poscnt[r];
      ns += negsum[r]; nc += (float)negcnt[r];
    }
    atomicAdd(&acc[0], loss_s);
    atomicAdd(&acc[1], err);
    atomicAdd(&acc[2], ps);
    atomicAdd(&acc[3], pc);
    atomicAdd(&acc[4], ns);
    atomicAdd(&acc[5], nc);
  }
}

// ---------------------------------------------------------------------------
// Kernel 3: fold the six accumulators into the 4 scalar outputs.
// ---------------------------------------------------------------------------
__global__ void finalize_kernel(const float* __restrict__ acc, float* __restrict__ out, int N) {
  if (threadIdx.x == 0 && blockIdx.x == 0) {
    float n = (float)N;
    out[0] = acc[0] / n;          // loss
    out[1] = 1.0f - acc[1] / n;   // prec
    out[2] = acc[2] / acc[3];     // pos_d
    out[3] = acc[4] / acc[5];     // neg_d
  }
}

extern "C" void kernel_launch(void* const* d_in, const int* in_sizes, int n_in,
                              void* d_out, int out_size, void* d_ws, size_t ws_size,
                              hipStream_t stream) {
  const float* X       = (const float*)d_in[0];
  const int*   targets = (const int*)d_in[1];
  const int N = in_sizes[1];
  const int D = in_sizes[0] / N;

  float* sq  = (float*)d_ws;     // [N]
  float* acc = sq + N;           // [8] accumulators
  float* out = (float*)d_out;    // 4 scalars

  sq_norm_kernel<<<(N + 255) / 256, 256, 0, stream>>>(X, sq, acc, N, D);

  // LDS: A panel + dist slab + column targets + per-row stat arrays
  size_t smem = (size_t)16 * (D + 4) * 4   // Abuf      (33,024 B @ D=512)
              + (size_t)16 * (N + 4) * 4   // distbuf   (262,400 B @ N=4096)
              + (size_t)N * 4              // tgtcol    (16,384 B)
              + 16 * 4 * 10;               // stat arrays (+slack)  -> ~312.4 KB of 320 KB
  if (D == 512) {
    neighbour_main<512><<<N / 16, 512, smem, stream>>>(X, targets, sq, acc, N, D);
  } else {
    neighbour_main<0><<<N / 16, 512, smem, stream>>>(X, targets, sq, acc, N, D);
  }

  finalize_kernel<<<1, 32, 0, stream>>>(acc, out, N);
}